// DXYNet_78443282694820
// MI455X (gfx1250) — compile-verified
//
#include <hip/hip_runtime.h>
#include <hip/hip_bf16.h>
#include <math.h>

// ---------------------------------------------------------------------------
// MI455X (gfx1250) implementation.
// Roofline: dominated by the 205 MB image_feats stream (segment mean); at
// 23.3 TB/s that's ~9 us. All C=1000-sized GEMMs (~3.6 GFLOP total) are
// compute-trivial -> exact f32 WMMA 16x16x4, branch-free inner loop.
// ---------------------------------------------------------------------------

typedef __attribute__((ext_vector_type(2))) float v2f;
typedef __attribute__((ext_vector_type(8))) float v8f;

#define T_TEMP   10.0f
#define THRESH   0.5f
#define NEG_BIG  (-9e15f)

// ---------------- wave32 block reductions (256 threads = 8 waves) ----------
__device__ __forceinline__ float blockReduceMax(float v, float* s) {
#pragma unroll
    for (int o = 16; o > 0; o >>= 1) v = fmaxf(v, __shfl_xor(v, o, 32));
    const int lane = threadIdx.x & 31, w = threadIdx.x >> 5;
    if (lane == 0) s[w] = v;
    __syncthreads();
    const int nw = blockDim.x >> 5;
    float r = s[0];
    for (int i = 1; i < nw; ++i) r = fmaxf(r, s[i]);
    __syncthreads();
    return r;
}

__device__ __forceinline__ float blockReduceSum(float v, float* s) {
#pragma unroll
    for (int o = 16; o > 0; o >>= 1) v += __shfl_xor(v, o, 32);
    const int lane = threadIdx.x & 31, w = threadIdx.x >> 5;
    if (lane == 0) s[w] = v;
    __syncthreads();
    const int nw = blockDim.x >> 5;
    float r = s[0];
    for (int i = 1; i < nw; ++i) r += s[i];
    __syncthreads();
    return r;
}

// ---------------- per-class counts -----------------------------------------
__global__ void count_kernel(const int* __restrict__ labels, float* __restrict__ counts, int N) {
    int i = blockIdx.x * blockDim.x + threadIdx.x;
    if (i < N) atomicAdd(&counts[labels[i]], 1.0f);
}

// ---------------- segment sum: the only HBM-heavy kernel -------------------
// Streams image_feats (N x D f32) with float4 loads; L2-resident atomics
// (sums buffer is 2 MB << 192 MB L2).
__global__ void segsum_kernel(const float* __restrict__ feats, const int* __restrict__ labels,
                              float* __restrict__ sums, int N, int D4) {
    long long idx    = (long long)blockIdx.x * blockDim.x + threadIdx.x;
    long long total  = (long long)N * D4;
    long long stride = (long long)gridDim.x * blockDim.x;
    for (; idx < total; idx += stride) {
        int row = (int)(idx / D4);
        int c4  = (int)(idx % D4);
        int lab = labels[row];
        float4 v = reinterpret_cast<const float4*>(feats)[idx];
        float* dst = sums + (long long)lab * D4 * 4 + c4 * 4;
        atomicAdd(dst + 0, v.x);
        atomicAdd(dst + 1, v.y);
        atomicAdd(dst + 2, v.z);
        atomicAdd(dst + 3, v.w);
    }
}

// ---------------- protos[c] = sums[tpl[c]] / counts[tpl[c]] ----------------
__global__ void protos_kernel(const float* __restrict__ sums, const float* __restrict__ counts,
                              const int* __restrict__ tpl, float* __restrict__ protos,
                              int C, int D) {
    int i = blockIdx.x * blockDim.x + threadIdx.x;
    if (i >= C * D) return;
    int c = i / D, j = i % D;
    int src = tpl[c];
    protos[i] = sums[(long long)src * D + j] / counts[src];
}

// ---------------- L2-row-normalize (one block per row) ---------------------
__global__ void rownorm_kernel(const float* __restrict__ in, float* __restrict__ out,
                               int rows, int D) {
    __shared__ float s[8];
    const int row = blockIdx.x;
    (void)rows;
    float ss = 0.f;
    for (int j = threadIdx.x; j < D; j += blockDim.x) {
        float v = in[(long long)row * D + j];
        ss += v * v;
    }
    ss = blockReduceSum(ss, s);
    float inv = 1.0f / sqrtf(ss);
    for (int j = threadIdx.x; j < D; j += blockDim.x)
        out[(long long)row * D + j] = in[(long long)row * D + j] * inv;
}

// ---------------- masked softmax: softmax((d>0.5 ? d : -9e15)*10) ----------
__global__ void softmax_kernel(float* __restrict__ Dm, int rows, int cols) {
    __shared__ float s[8];
    const int row = blockIdx.x;
    (void)rows;
    float* r = Dm + (long long)row * cols;

    float mx = -INFINITY;
    for (int j = threadIdx.x; j < cols; j += blockDim.x) {
        float d = r[j];
        float v = (d > THRESH ? d : NEG_BIG) * T_TEMP;
        mx = fmaxf(mx, v);
    }
    mx = blockReduceMax(mx, s);

    float sum = 0.f;
    for (int j = threadIdx.x; j < cols; j += blockDim.x) {
        float d = r[j];
        float v = (d > THRESH ? d : NEG_BIG) * T_TEMP;
        float e = expf(v - mx);
        r[j] = e;
        sum += e;
    }
    sum = blockReduceSum(sum, s);

    float inv = 1.0f / sum;
    for (int j = threadIdx.x; j < cols; j += blockDim.x)
        r[j] *= inv;
}

// ---------------- f32 WMMA GEMM: C[M,N] = A[M,K] @ op(B) -------------------
// One 16x16 output tile per wave32; K stepped by 4 via V_WMMA_F32_16X16X4_F32.
// ISA VGPR layouts (CDNA5 ISA):
//   A 16x4 : lanes 0-15 -> M=lane, K={0,1}; lanes 16-31 -> M=lane-16, K={2,3}
//   B  4x16: VGPR0 holds K rows {0 | 2}, VGPR1 holds K rows {1 | 3} across halves
//   C 16x16: VGPR i, lanes 0-15 -> M=i, lanes 16-31 -> M=i+8; N=lane&15
//
// Out-of-range rows/cols are CLAMPED (not zero-filled): each output element
// (m,n) depends only on A row m / B col n, and the guarded store drops every
// padded element, so clamping is exact and keeps the inner loop branch-free
// (unconditional global_load_b64 per operand, EXEC all-1s throughout).
template <bool TRANSB>
__global__ void wmma_gemm_kernel(const float* __restrict__ A, const float* __restrict__ B,
                                 float* __restrict__ Cm,
                                 int M, int N, int K, int lda, int ldb, int ldc) {
    const int wave = threadIdx.x >> 5;
    const int lane = threadIdx.x & 31;
    const int half = lane >> 4;     // 0: K offset {0,1} ; 1: K offset {2,3}
    const int l    = lane & 15;

    const int tilesN = (N + 15) >> 4;
    const int tilesM = (M + 15) >> 4;
    const int tile   = blockIdx.x * (blockDim.x >> 5) + wave;
    if (tile >= tilesM * tilesN) return;   // wave-uniform: EXEC stays all-1s

    const int tm = tile / tilesN;
    const int tn = tile % tilesN;
    const int aRow = tm * 16 + l;
    const int bCol = tn * 16 + l;
    const int aRowC = aRow < M ? aRow : (M - 1);   // clamp: store guard covers it
    const int bColC = bCol < N ? bCol : (N - 1);

    // per-lane base pointers (kk = half*2 start), stepped by 4 floats per iter
    const float* aPtr = A + (long long)aRowC * lda + half * 2;
    const float* bPtr;
    long long bStep;
    if (TRANSB) { // B element (k,n) = Bsrc[n*ldb + k]  (Gram matrix path)
        bPtr  = B + (long long)bColC * ldb + half * 2;
        bStep = 4;
    } else {
        bPtr  = B + (long long)(half * 2) * ldb + bColC;
        bStep = 4LL * ldb;
    }

    v8f acc = {};
    for (int k = 0; k < K; k += 4) {   // all K here are multiples of 4
        v2f a = *reinterpret_cast<const v2f*>(aPtr);   // K-consecutive pair: b64 load
        v2f b;
        if (TRANSB) {
            b = *reinterpret_cast<const v2f*>(bPtr);
        } else {
            b.x = bPtr[0];
            b.y = bPtr[ldb];
        }
        acc = __builtin_amdgcn_wmma_f32_16x16x4_f32(
            /*neg_a=*/false, a, /*neg_b=*/false, b,
            /*c_mod=*/(short)0, acc, /*reuse_a=*/false, /*reuse_b=*/false);
        aPtr += 4;
        bPtr += bStep;
    }

    if (bCol < N) {
#pragma unroll
        for (int i = 0; i < 8; ++i) {
            int r = tm * 16 + i + half * 8;
            if (r < M) Cm[(long long)r * ldc + bCol] = acc[i];
        }
    }
}

// ---------------------------------------------------------------------------
extern "C" void kernel_launch(void* const* d_in, const int* in_sizes, int n_in,
                              void* d_out, int out_size, void* d_ws, size_t ws_size,
                              hipStream_t stream) {
    (void)n_in; (void)out_size; (void)ws_size;

    const float* image_feats = (const float*)d_in[0];  // [N, IMG]
    const float* attributes  = (const float*)d_in[1];  // [C, ATT]
    const float* att_g       = (const float*)d_in[2];  // [ATT, H]
    const float* att_h       = (const float*)d_in[3];  // [ATT, H]
    const int*   labels      = (const int*)d_in[4];    // [N]
    const int*   tpl         = (const int*)d_in[5];    // [C]

    const int N   = in_sizes[4];        // 100000
    const int C   = in_sizes[5];        // 1000
    const int IMG = in_sizes[0] / N;    // 512
    const int ATT = in_sizes[1] / C;    // 512
    const int H   = in_sizes[2] / ATT;  // 256
    float* out = (float*)d_out;         // [C, IMG]

    // ---- workspace layout (256B aligned slabs, ~12.3 MB total) ----
    char* ws = (char*)d_ws;
    auto alloc = [&](size_t bytes) -> float* {
        char* p = ws;
        ws += (bytes + 255) & ~(size_t)255;
        return (float*)p;
    };
    float* sums   = alloc((size_t)C * IMG * 4);  // segment sums
    float* counts = alloc((size_t)C * 4);
    float* protos = alloc((size_t)C * IMG * 4);
    float* P      = alloc((size_t)C * H * 4);    // projection (reused for q)
    float* Pn     = alloc((size_t)C * H * 4);    // normalized rows
    float* Dm     = alloc((size_t)C * C * 4);    // cosine / attention matrix
    float* enc    = alloc((size_t)C * ATT * 4);  // encoded attributes

    // ---- phase 1: class prototypes (HBM-bound part) ----
    hipMemsetAsync(sums,   0, (size_t)C * IMG * 4, stream);
    hipMemsetAsync(counts, 0, (size_t)C * 4,       stream);
    count_kernel<<<(N + 255) / 256, 256, 0, stream>>>(labels, counts, N);
    segsum_kernel<<<4096, 256, 0, stream>>>(image_feats, labels, sums, N, IMG / 4);
    protos_kernel<<<(C * IMG + 255) / 256, 256, 0, stream>>>(sums, counts, tpl, protos, C, IMG);

    auto gemm = [&](const float* A, const float* B, float* Cc,
                    int M_, int N_, int K_, int lda, int ldb, int ldc, bool transb) {
        int tiles  = ((M_ + 15) / 16) * ((N_ + 15) / 16);
        int blocks = (tiles + 7) / 8;  // 8 waves / 256-thread block (fills a WGP)
        if (transb)
            wmma_gemm_kernel<true ><<<blocks, 256, 0, stream>>>(A, B, Cc, M_, N_, K_, lda, ldb, ldc);
        else
            wmma_gemm_kernel<false><<<blocks, 256, 0, stream>>>(A, B, Cc, M_, N_, K_, lda, ldb, ldc);
    };

    // ---- phase 2: _encode_attr ----
    gemm(attributes, att_h, P, C, H, ATT, ATT, H, H, false);      // P = attrs @ att_h
    rownorm_kernel<<<C, 256, 0, stream>>>(P, Pn, C, H);
    gemm(Pn, Pn, Dm, C, C, H, H, H, C, true);                     // D = Pn @ Pn^T
    softmax_kernel<<<C, 256, 0, stream>>>(Dm, C, C);              // masked softmax
    gemm(Dm, attributes, enc, C, ATT, C, C, ATT, ATT, false);     // enc = W @ attrs

    // ---- phase 3: get_attention ----
    gemm(enc, att_g, P, C, H, ATT, ATT, H, H, false);             // q = enc @ att_g
    rownorm_kernel<<<C, 256, 0, stream>>>(P, Pn, C, H);
    gemm(Pn, Pn, Dm, C, C, H, H, H, C, true);                     // D2 = qn @ qn^T
    softmax_kernel<<<C, 256, 0, stream>>>(Dm, C, C);              // attention

    // ---- phase 4: output ----
    gemm(Dm, protos, out, C, IMG, C, C, IMG, IMG, false);         // attention @ protos
}